// PositionPassingTGN_39977555591654
// MI455X (gfx1250) — compile-verified
//
#include <hip/hip_runtime.h>

// ---------------------------------------------------------------------------
// PositionPassingTGN gather:
//   out[0            .. 64,000,000)   = memory_bank[n_id]       (500K x 128 f32)
//   out[64,000,000   .. 128,000,000)  = pos_memory_bank[n_id]   (500K x 128 f32)
//   out[128,000,000  .. 128,500,000)  = float(last_update_bank[n_id])
//
// Pure HBM-bandwidth problem (~1.03 GB total -> ~44us at 23.3 TB/s).
// CDNA5 path: async Global->LDS->Global DMA (ASYNCcnt), one 512B row per wave,
// 32 lanes x B128, bypassing the VGPR file.
// ---------------------------------------------------------------------------

#define N_IDS_C      500000
#define MEMORY_DIM_C 128

constexpr int WAVE_SZ       = 32;
constexpr int BLOCK_SZ      = 256;   // 8 waves per block
constexpr int ROWS_PER_WAVE = 4;     // 4 x 512B rows staged per wave per pass

#if __has_builtin(__builtin_amdgcn_global_load_async_to_lds_b128) && \
    __has_builtin(__builtin_amdgcn_global_store_async_from_lds_b128)
#define USE_ASYNC_LDS 1
#else
#define USE_ASYNC_LDS 0
#endif

#if __has_builtin(__builtin_amdgcn_s_wait_asynccnt)
#define WAIT_ASYNCCNT(n) __builtin_amdgcn_s_wait_asynccnt(n)
#else
#define WAIT_ASYNCCNT(n) asm volatile("s_wait_asynccnt %0" ::"i"(n) : "memory")
#endif

// b128 async ops take pointers to 4-int vectors, per hipcc's prototype
// diagnostic: param 0 is 'int __vector(4) __device__ *' (global AS).
typedef int v4i __attribute__((ext_vector_type(4)));
typedef __attribute__((address_space(1))) v4i* gptr_b128;
typedef __attribute__((address_space(3))) v4i* lptr_b128;
typedef __attribute__((address_space(1))) void gbl_void_t;
typedef __attribute__((address_space(3))) void lds_void_t;

__global__ __launch_bounds__(BLOCK_SZ) void tgn_gather_rows_kernel(
    const int* __restrict__ n_id,
    const float* __restrict__ memory_bank,
    const float* __restrict__ pos_memory_bank,
    float* __restrict__ out)
{
    const int  lane    = threadIdx.x & (WAVE_SZ - 1);
    const int  wib     = threadIdx.x >> 5;                       // wave in block
    const long wave_id = (long)blockIdx.x * (BLOCK_SZ / WAVE_SZ) + wib;
    const long base_r  = wave_id * ROWS_PER_WAVE;
    const long total_r = 2L * N_IDS_C;                           // z rows then pos_z rows

#if USE_ASYNC_LDS
    __shared__ float4 stage[ROWS_PER_WAVE][BLOCK_SZ];            // 16 KB / block

    // Issue ROWS_PER_WAVE async row-gathers (global -> LDS), each row is
    // 32 lanes x 16B fully coalesced.
    for (int k = 0; k < ROWS_PER_WAVE; ++k) {
        const long r = base_r + k;
        if (r >= total_r) break;                                 // wave-uniform
        const long rr = (r < N_IDS_C) ? r : (r - N_IDS_C);
        const float* __restrict__ src =
            (r < N_IDS_C) ? memory_bank : pos_memory_bank;
        const long idx = (long)n_id[rr];
        const float* g = src + idx * MEMORY_DIM_C + lane * 4;
        __builtin_amdgcn_global_load_async_to_lds_b128(
            (gptr_b128)(gbl_void_t*)g,
            (lptr_b128)(lds_void_t*)&stage[k][threadIdx.x],
            /*offset=*/0, /*cpol=*/0);
    }

    WAIT_ASYNCCNT(0);   // drain async loads before reading LDS via async stores

    // Async stores (LDS -> global): output rows are contiguous at r*128.
    for (int k = 0; k < ROWS_PER_WAVE; ++k) {
        const long r = base_r + k;
        if (r >= total_r) break;
        float* g = out + r * (long)MEMORY_DIM_C + lane * 4;
        __builtin_amdgcn_global_store_async_from_lds_b128(
            (gptr_b128)(gbl_void_t*)g,
            (lptr_b128)(lds_void_t*)&stage[k][threadIdx.x],
            /*offset=*/0, /*cpol=*/0);
    }
    WAIT_ASYNCCNT(0);
#else
    // Fallback: direct VGPR path, still one 512B row per wave via float4.
    for (int k = 0; k < ROWS_PER_WAVE; ++k) {
        const long r = base_r + k;
        if (r >= total_r) break;
        const long rr = (r < N_IDS_C) ? r : (r - N_IDS_C);
        const float* __restrict__ src =
            (r < N_IDS_C) ? memory_bank : pos_memory_bank;
        const long idx = (long)n_id[rr];
        const float4 v =
            *(reinterpret_cast<const float4*>(src + idx * MEMORY_DIM_C) + lane);
        *(reinterpret_cast<float4*>(out + r * (long)MEMORY_DIM_C) + lane) = v;
    }
#endif
}

__global__ __launch_bounds__(BLOCK_SZ) void tgn_last_update_kernel(
    const int* __restrict__ n_id,
    const int* __restrict__ last_update_bank,
    float* __restrict__ out_lu)
{
    const int i = blockIdx.x * blockDim.x + threadIdx.x;
    if (i < N_IDS_C) {
        out_lu[i] = (float)last_update_bank[n_id[i]];
    }
}

extern "C" void kernel_launch(void* const* d_in, const int* in_sizes, int n_in,
                              void* d_out, int out_size, void* d_ws, size_t ws_size,
                              hipStream_t stream)
{
    (void)in_sizes; (void)n_in; (void)out_size; (void)d_ws; (void)ws_size;

    const int*   n_id    = (const int*)d_in[0];
    const float* mem     = (const float*)d_in[1];
    const float* pos_mem = (const float*)d_in[2];
    const int*   lu      = (const int*)d_in[3];
    float*       out     = (float*)d_out;

    // 2*N rows, ROWS_PER_WAVE rows per wave, 8 waves per 256-thread block.
    const long total_rows = 2L * N_IDS_C;
    const long waves   = (total_rows + ROWS_PER_WAVE - 1) / ROWS_PER_WAVE;
    const long threads = waves * WAVE_SZ;
    const int  blocks  = (int)((threads + BLOCK_SZ - 1) / BLOCK_SZ);

    tgn_gather_rows_kernel<<<blocks, BLOCK_SZ, 0, stream>>>(n_id, mem, pos_mem, out);

    float* out_lu = out + total_rows * (long)MEMORY_DIM_C;   // offset 128,000,000
    const int lu_blocks = (N_IDS_C + BLOCK_SZ - 1) / BLOCK_SZ;
    tgn_last_update_kernel<<<lu_blocks, BLOCK_SZ, 0, stream>>>(n_id, lu, out_lu);
}